// graph_convolution_55387898249725
// MI455X (gfx1250) — compile-verified
//
#include <hip/hip_runtime.h>
#include <hip/hip_bf16.h>

typedef __attribute__((ext_vector_type(2))) float v2f;
typedef __attribute__((ext_vector_type(8))) float v8f;

#define B_   64
#define FIN  128
#define NN   22
#define SS   50
#define FOUT 128
#define HH   4
#define D1   (NN * FIN)   // 2816
#define D2   (SS * FOUT)  // 6400

// ---------------- K1 (WMMA): lr1[b,slot,s] = sum_d awt[slot,d] * y1[b,s,d] ----------
// y1[b,s,d] = x[b, d&127, d>>7, s].  Per b: [8x2816] . [2816x50] GEMM.
// A rows 8..15 are don't-care (their D rows are never stored; no cross-row mixing in
// D=A*B), so lanes l16>=8 just wrap to a valid awt address -> all loads unconditional.
__global__ void k1_lr1(const float* __restrict__ x, const float* __restrict__ awt,
                       float* __restrict__ lr1) {
  const int lane = threadIdx.x & 31;
  const int wid = blockIdx.x * (blockDim.x >> 5) + (threadIdx.x >> 5);
  const int b = wid >> 2, st = wid & 3;
  const int half = lane >> 4, l16 = lane & 15;
  int s = st * 16 + l16;
  if (s > SS - 1) s = SS - 1;
  const float* xb = x + (size_t)b * FIN * NN * SS;
  const int hsel = (l16 >> 1) & 3, lrsel = l16 & 1;  // wraps for l16>=8 (don't-care rows)
  const float* aptr = awt + hsel * (2 * D1) + lrsel * D1;  // + d
  v8f acc = {};
  for (int d0 = 0; d0 < D1; d0 += 4) {
    const int dk = d0 + 2 * half;
    v2f a, bm;
    a.x = aptr[dk + 0];
    a.y = aptr[dk + 1];
    const int dA = dk, dB = dk + 1;
    bm.x = xb[(size_t)(dA & 127) * (NN * SS) + (dA >> 7) * SS + s];
    bm.y = xb[(size_t)(dB & 127) * (NN * SS) + (dB >> 7) * SS + s];
    acc = __builtin_amdgcn_wmma_f32_16x16x4_f32(false, a, false, bm, (short)0, acc,
                                                false, false);
  }
  // D[lane][r] = C[M=r+8*half][N=l16]; valid slots are M=0..7 -> half==0 lanes only
  if (half == 0) {
    const int sg = st * 16 + l16;
#pragma unroll
    for (int r = 0; r < 8; ++r) {
      if (sg < SS) lr1[(b * 8 + r) * SS + sg] = acc[r];
    }
  }
}

// ---------------- K2: W1[b,i,j] = mean_h softmax_j(lrelu(l+r)) ----------------
__global__ void k2_w1(const float* __restrict__ lr1, float* __restrict__ W1) {
  const int b = blockIdx.x, tid = threadIdx.x;
  __shared__ float L[HH][SS], R[HH][SS];
  if (tid < SS) {
#pragma unroll
    for (int h = 0; h < HH; ++h) {
      L[h][tid] = lr1[b * 8 * SS + (2 * h + 0) * SS + tid];
      R[h][tid] = lr1[b * 8 * SS + (2 * h + 1) * SS + tid];
    }
  }
  __syncthreads();
  if (tid < SS) {
    float w[SS];
#pragma unroll
    for (int j = 0; j < SS; ++j) w[j] = 0.f;
#pragma unroll
    for (int h = 0; h < HH; ++h) {
      const float li = L[h][tid];
      float e[SS], m = -3.4e38f;
#pragma unroll
      for (int j = 0; j < SS; ++j) {
        float v = li + R[h][j];
        v = v > 0.f ? v : 0.2f * v;
        e[j] = v;
        m = fmaxf(m, v);
      }
      float sum = 0.f;
#pragma unroll
      for (int j = 0; j < SS; ++j) { e[j] = __expf(e[j] - m); sum += e[j]; }
      const float inv = 0.25f / sum;
#pragma unroll
      for (int j = 0; j < SS; ++j) w[j] += e[j] * inv;
    }
#pragma unroll
    for (int j = 0; j < SS; ++j) W1[(b * SS + tid) * SS + j] = w[j];
  }
}

// ---------------- K3 (WMMA): yf[b,s,n,fo] = sum_f x[b,f,n,s] * fw[f,fo] --------------
__global__ void k3_feat(const float* __restrict__ x, const float* __restrict__ fw,
                        float* __restrict__ yf) {
  const int lane = threadIdx.x & 31;
  const int wid = blockIdx.x * (blockDim.x >> 5) + (threadIdx.x >> 5);
  const int perB = NN * 4 * 8;  // 22 n * 4 s-tiles * 8 fo-tiles = 704
  const int b = wid / perB;
  const int t = wid % perB;
  const int n = t >> 5, st = (t >> 3) & 3, tn = t & 7;
  const int half = lane >> 4, l16 = lane & 15;
  int s = st * 16 + l16;
  if (s > SS - 1) s = SS - 1;
  const float* arow = x + (size_t)b * FIN * NN * SS + n * SS + s;  // + k*1100
  const int fo = tn * 16 + l16;
  v8f acc = {};
  for (int k0 = 0; k0 < FIN; k0 += 4) {
    const int ka = k0 + 2 * half;
    v2f a, bm;
    a.x = arow[(size_t)(ka + 0) * (NN * SS)];
    a.y = arow[(size_t)(ka + 1) * (NN * SS)];
    bm.x = fw[(ka + 0) * FOUT + fo];
    bm.y = fw[(ka + 1) * FOUT + fo];
    acc = __builtin_amdgcn_wmma_f32_16x16x4_f32(false, a, false, bm, (short)0, acc,
                                                false, false);
  }
#pragma unroll
  for (int r = 0; r < 8; ++r) {
    const int sg = st * 16 + r + 8 * half;
    if (sg < SS) yf[(((size_t)b * SS + sg) * NN + n) * FOUT + fo] = acc[r];
  }
}

// ---------------- K4 (WMMA): z[b,s,c] = sum_j W1[b,s,j] * yf[b,j,c] -------------------
// Main loop fully in-bounds (K=0..47); peeled tail zero-pads K=50,51 via cndmask.
__global__ void k4_attn1(const float* __restrict__ W1, const float* __restrict__ yf,
                         float* __restrict__ z) {
  const int lane = threadIdx.x & 31;
  const int wid = blockIdx.x * (blockDim.x >> 5) + (threadIdx.x >> 5);
  const int tilesN = D1 / 16;  // 176
  const int perB = 4 * tilesN;
  const int b = wid / perB, t = wid % perB;
  const int tm = t / tilesN, tn = t % tilesN;
  const int half = lane >> 4, l16 = lane & 15;
  const int m = tm * 16 + l16;
  const int mc = m < SS ? m : SS - 1;
  const float* Wb = W1 + (size_t)b * SS * SS;
  const float* yfb = yf + (size_t)b * SS * D1;
  const int c = tn * 16 + l16;
  v8f acc = {};
  for (int k0 = 0; k0 < 48; k0 += 4) {
    const int ka = k0 + 2 * half;
    v2f a, bm;
    a.x = Wb[mc * SS + ka + 0];
    a.y = Wb[mc * SS + ka + 1];
    bm.x = yfb[(size_t)(ka + 0) * D1 + c];
    bm.y = yfb[(size_t)(ka + 1) * D1 + c];
    acc = __builtin_amdgcn_wmma_f32_16x16x4_f32(false, a, false, bm, (short)0, acc,
                                                false, false);
  }
  {  // tail: ka = 48,49 (half==0, valid) or 50,51 (half==1, zero-pad A)
    const int ka = 48 + 2 * half;
    const int k0c = (ka + 0 < SS) ? ka + 0 : SS - 1;
    const int k1c = (ka + 1 < SS) ? ka + 1 : SS - 1;
    const float w0 = Wb[mc * SS + k0c];
    const float w1 = Wb[mc * SS + k1c];
    v2f a, bm;
    a.x = (ka + 0 < SS) ? w0 : 0.f;
    a.y = (ka + 1 < SS) ? w1 : 0.f;
    bm.x = yfb[(size_t)k0c * D1 + c];
    bm.y = yfb[(size_t)k1c * D1 + c];
    acc = __builtin_amdgcn_wmma_f32_16x16x4_f32(false, a, false, bm, (short)0, acc,
                                                false, false);
  }
#pragma unroll
  for (int r = 0; r < 8; ++r) {
    const int mg = tm * 16 + r + 8 * half;
    if (mg < SS) z[((size_t)b * SS + mg) * D1 + c] = acc[r];
  }
}

// ---------------- K5: stage-2 left/right projections (coalesced in fo) ----------------
__global__ void k5_lr2(const float* __restrict__ z, const float* __restrict__ aw,
                       float* __restrict__ lr2) {
  const int n = blockIdx.x, b = blockIdx.y, tid = threadIdx.x;
  float acc[8];
#pragma unroll
  for (int i = 0; i < 8; ++i) acc[i] = 0.f;
  for (int d = tid; d < D2; d += 256) {
    const int fo = d & 127, s = d >> 7;
    const float zv = z[(((size_t)b * SS + s) * NN + n) * FOUT + fo];
#pragma unroll
    for (int h = 0; h < HH; ++h) {
      acc[2 * h + 0] += zv * aw[h * (2 * D2) + d];
      acc[2 * h + 1] += zv * aw[h * (2 * D2) + D2 + d];
    }
  }
  __shared__ float red[8][256];
#pragma unroll
  for (int i = 0; i < 8; ++i) red[i][tid] = acc[i];
  __syncthreads();
  for (int off = 128; off > 0; off >>= 1) {
    if (tid < off) {
#pragma unroll
      for (int i = 0; i < 8; ++i) red[i][tid] += red[i][tid + off];
    }
    __syncthreads();
  }
  if (tid < 8) lr2[(b * 8 + tid) * NN + n] = red[tid][0];
}

// ---------------- K5b: W2[b,i,j] over N=22 ----------------
__global__ void k5b_w2(const float* __restrict__ lr2, float* __restrict__ W2) {
  const int b = blockIdx.x, tid = threadIdx.x;
  __shared__ float L[HH][NN], R[HH][NN];
  if (tid < NN) {
#pragma unroll
    for (int h = 0; h < HH; ++h) {
      L[h][tid] = lr2[b * 8 * NN + (2 * h + 0) * NN + tid];
      R[h][tid] = lr2[b * 8 * NN + (2 * h + 1) * NN + tid];
    }
  }
  __syncthreads();
  if (tid < NN) {
    float w[NN];
#pragma unroll
    for (int j = 0; j < NN; ++j) w[j] = 0.f;
#pragma unroll
    for (int h = 0; h < HH; ++h) {
      const float li = L[h][tid];
      float e[NN], m = -3.4e38f;
#pragma unroll
      for (int j = 0; j < NN; ++j) {
        float v = li + R[h][j];
        v = v > 0.f ? v : 0.2f * v;
        e[j] = v;
        m = fmaxf(m, v);
      }
      float sum = 0.f;
#pragma unroll
      for (int j = 0; j < NN; ++j) { e[j] = __expf(e[j] - m); sum += e[j]; }
      const float inv = 0.25f / sum;
#pragma unroll
      for (int j = 0; j < NN; ++j) w[j] += e[j] * inv;
    }
#pragma unroll
    for (int j = 0; j < NN; ++j) W2[(b * NN + tid) * NN + j] = w[j];
  }
}

// ---------------- K6 (WMMA): out[b,fo,n,s] = sum_j W2[b,n,j]*z[b,s,j,fo] + bias[s] ----
// Main loop in-bounds (K=0..19); peeled tail zero-pads K=22,23 via cndmask.
__global__ void k6_out(const float* __restrict__ W2, const float* __restrict__ z,
                       const float* __restrict__ bias, float* __restrict__ out) {
  const int lane = threadIdx.x & 31;
  const int wid = blockIdx.x * (blockDim.x >> 5) + (threadIdx.x >> 5);
  const int tilesN = FOUT / 16;  // 8
  const int perBS = 2 * tilesN;  // M=22 -> 2 tiles
  const int bs = wid / perBS, t = wid % perBS;
  const int b = bs / SS, s = bs % SS;
  const int tm = t / tilesN, tn = t % tilesN;
  const int half = lane >> 4, l16 = lane & 15;
  const int m = tm * 16 + l16;
  const int mc = m < NN ? m : NN - 1;
  const float* Wb = W2 + (size_t)b * NN * NN;
  const float* zb = z + ((size_t)b * SS + s) * NN * FOUT;
  const int fo = tn * 16 + l16;
  v8f acc = {};
  for (int k0 = 0; k0 < 20; k0 += 4) {
    const int ka = k0 + 2 * half;
    v2f a, bm;
    a.x = Wb[mc * NN + ka + 0];
    a.y = Wb[mc * NN + ka + 1];
    bm.x = zb[(ka + 0) * FOUT + fo];
    bm.y = zb[(ka + 1) * FOUT + fo];
    acc = __builtin_amdgcn_wmma_f32_16x16x4_f32(false, a, false, bm, (short)0, acc,
                                                false, false);
  }
  {  // tail: ka = 20,21 (half==0, valid) or 22,23 (half==1, zero-pad A)
    const int ka = 20 + 2 * half;
    const int k0c = (ka + 0 < NN) ? ka + 0 : NN - 1;
    const int k1c = (ka + 1 < NN) ? ka + 1 : NN - 1;
    const float w0 = Wb[mc * NN + k0c];
    const float w1 = Wb[mc * NN + k1c];
    v2f a, bm;
    a.x = (ka + 0 < NN) ? w0 : 0.f;
    a.y = (ka + 1 < NN) ? w1 : 0.f;
    bm.x = zb[k0c * FOUT + fo];
    bm.y = zb[k1c * FOUT + fo];
    acc = __builtin_amdgcn_wmma_f32_16x16x4_f32(false, a, false, bm, (short)0, acc,
                                                false, false);
  }
  const float bv = bias[s];
#pragma unroll
  for (int r = 0; r < 8; ++r) {
    const int ng = tm * 16 + r + 8 * half;
    if (ng < NN) out[(((size_t)b * FOUT + fo) * NN + ng) * SS + s] = acc[r] + bv;
  }
}

extern "C" void kernel_launch(void* const* d_in, const int* in_sizes, int n_in,
                              void* d_out, int out_size, void* d_ws, size_t ws_size,
                              hipStream_t stream) {
  (void)in_sizes; (void)n_in; (void)out_size; (void)ws_size;
  const float* x    = (const float*)d_in[0];
  const float* awt  = (const float*)d_in[1];
  const float* fw   = (const float*)d_in[2];
  const float* aw   = (const float*)d_in[3];
  const float* bias = (const float*)d_in[4];
  float* out = (float*)d_out;

  float* ws  = (float*)d_ws;
  float* lr1 = ws;                        // 64*8*50   = 25600
  float* W1  = lr1 + (size_t)B_ * 8 * SS; // 64*50*50  = 160000
  float* lr2 = W1 + (size_t)B_ * SS * SS; // 64*8*22   = 11264
  float* W2  = lr2 + (size_t)B_ * 8 * NN; // 64*22*22  = 30976
  float* yf  = W2 + (size_t)B_ * NN * NN; // 9,011,200
  float* z   = yf + (size_t)B_ * SS * NN * FOUT; // 9,011,200

  // Stage 1 attention weights (WMMA GEMM, coalesced x reads)
  k1_lr1<<<32, 256, 0, stream>>>(x, awt, lr1);        // 64*4 waves
  k2_w1<<<B_, 64, 0, stream>>>(lr1, W1);
  // Feature projection (x @ fw); commutes with W1 mixing
  k3_feat<<<5632, 256, 0, stream>>>(x, fw, yf);       // 64*704 waves
  // Stage-1 aggregation: z = W1 @ yf
  k4_attn1<<<5632, 256, 0, stream>>>(W1, yf, z);      // 64*4*176 waves
  // Stage 2 attention weights
  k5_lr2<<<dim3(NN, B_), 256, 0, stream>>>(z, aw, lr2);
  k5b_w2<<<B_, 32, 0, stream>>>(lr2, W2);
  // Stage-2 aggregation + transpose + bias
  k6_out<<<6400, 256, 0, stream>>>(W2, z, bias, out); // 64*50*16 waves
}